// RGBDConsistency_61795989455186
// MI455X (gfx1250) — compile-verified
//
#include <hip/hip_runtime.h>
#include <hip/hip_bf16.h>

#define Hdim 512
#define Wdim 512
#define NPIX (Hdim * Wdim)        // 262144
#define BS 16
#define BLOCK 256
#define PIXBLOCKS (NPIX / BLOCK)  // 1024 blocks per (batch,dir)
#define CHUNKS_PER_BATCH (2 * PIXBLOCKS)  // 2048 partials per batch (both dirs)
#define CLAMP_T 0.1f

typedef __attribute__((ext_vector_type(2))) float v2f;
typedef __attribute__((ext_vector_type(8))) float v8f;

// ---------------------------------------------------------------------------
// Kernel 1: per-batch geometry collapse.
// uvw(x,y,d) = d * ((x,y,1) @ P_b) + c_b   with
//   A  = R_src @ R_dst^T @ K^T
//   P  = Ki^T @ A
//   c  = t_dst @ K^T - t_src @ A
// One thread per (batch); both directions computed. 16x2x12 floats to ws.
// ---------------------------------------------------------------------------
__global__ void rgbd_precompute(const float* __restrict__ R0g, const float* __restrict__ t0g,
                                const float* __restrict__ R1g, const float* __restrict__ t1g,
                                const float* __restrict__ Kg, float* __restrict__ params) {
    int b = threadIdx.x;
    if (b >= BS) return;

    float K[3][3], Ki[3][3];
    for (int i = 0; i < 3; i++)
        for (int j = 0; j < 3; j++) K[i][j] = Kg[i * 3 + j];

    float det = K[0][0] * (K[1][1] * K[2][2] - K[1][2] * K[2][1])
              - K[0][1] * (K[1][0] * K[2][2] - K[1][2] * K[2][0])
              + K[0][2] * (K[1][0] * K[2][1] - K[1][1] * K[2][0]);
    float inv = 1.0f / det;
    Ki[0][0] =  (K[1][1] * K[2][2] - K[1][2] * K[2][1]) * inv;
    Ki[0][1] = -(K[0][1] * K[2][2] - K[0][2] * K[2][1]) * inv;
    Ki[0][2] =  (K[0][1] * K[1][2] - K[0][2] * K[1][1]) * inv;
    Ki[1][0] = -(K[1][0] * K[2][2] - K[1][2] * K[2][0]) * inv;
    Ki[1][1] =  (K[0][0] * K[2][2] - K[0][2] * K[2][0]) * inv;
    Ki[1][2] = -(K[0][0] * K[1][2] - K[0][2] * K[1][0]) * inv;
    Ki[2][0] =  (K[1][0] * K[2][1] - K[1][1] * K[2][0]) * inv;
    Ki[2][1] = -(K[0][0] * K[2][1] - K[0][1] * K[2][0]) * inv;
    Ki[2][2] =  (K[0][0] * K[1][1] - K[0][1] * K[1][0]) * inv;

    for (int dir = 0; dir < 2; dir++) {
        const float* Rs = (dir == 0) ? R0g : R1g;
        const float* Rd = (dir == 0) ? R1g : R0g;
        const float* ts = (dir == 0) ? t0g : t1g;
        const float* td = (dir == 0) ? t1g : t0g;
        float Ra[3][3], Rb[3][3], ta[3], tb[3];
        for (int i = 0; i < 3; i++) {
            ta[i] = ts[b * 3 + i];
            tb[i] = td[b * 3 + i];
            for (int j = 0; j < 3; j++) {
                Ra[i][j] = Rs[b * 9 + i * 3 + j];
                Rb[i][j] = Rd[b * 9 + i * 3 + j];
            }
        }
        float T[3][3], A[3][3], P[3][3], c[3];
        for (int i = 0; i < 3; i++)
            for (int k = 0; k < 3; k++) {
                float s = 0.f;
                for (int j = 0; j < 3; j++) s += Ra[i][j] * Rb[k][j];  // R_src @ R_dst^T
                T[i][k] = s;
            }
        for (int i = 0; i < 3; i++)
            for (int k = 0; k < 3; k++) {
                float s = 0.f;
                for (int j = 0; j < 3; j++) s += T[i][j] * K[k][j];    // @ K^T
                A[i][k] = s;
            }
        for (int k = 0; k < 3; k++) {
            float s1 = 0.f, s2 = 0.f;
            for (int j = 0; j < 3; j++) { s1 += tb[j] * K[k][j]; s2 += ta[j] * A[j][k]; }
            c[k] = s1 - s2;
        }
        for (int i = 0; i < 3; i++)
            for (int k = 0; k < 3; k++) {
                float s = 0.f;
                for (int j = 0; j < 3; j++) s += Ki[j][i] * A[j][k];   // Ki^T @ A
                P[i][k] = s;
            }
        float* o = params + (dir * BS + b) * 12;
        for (int i = 0; i < 3; i++)
            for (int j = 0; j < 3; j++) o[i * 3 + j] = P[i][j];
        o[9] = c[0]; o[10] = c[1]; o[11] = c[2];
    }
}

// ---------------------------------------------------------------------------
// Kernel 2: per-pixel reprojection + bilinear gather + masked/clamped L1,
// block-level deterministic reduction -> per-block partial sums.
// grid = (1024, 16, 2), block = 256 (8 wave32s).
// ---------------------------------------------------------------------------
__global__ __launch_bounds__(BLOCK)
void rgbd_reproject_loss(const float* __restrict__ depth0, const float* __restrict__ depth1,
                         const float* __restrict__ rgb0, const float* __restrict__ rgb1,
                         const unsigned char* __restrict__ mask0,
                         const unsigned char* __restrict__ mask1,
                         const float* __restrict__ params,
                         float* __restrict__ pd, float* __restrict__ prgb) {
    const int n   = blockIdx.x * BLOCK + threadIdx.x;
    const int b   = blockIdx.y;
    const int dir = blockIdx.z;

    const float* dsrc = (dir == 0) ? depth0 : depth1;
    const float* ddst = (dir == 0) ? depth1 : depth0;
    const float* csrc = (dir == 0) ? rgb0 : rgb1;
    const float* cdst = (dir == 0) ? rgb1 : rgb0;
    const unsigned char* msrc = (dir == 0) ? mask0 : mask1;

    const float* p = params + (dir * BS + b) * 12;
    const float P00 = p[0], P01 = p[1], P02 = p[2];
    const float P10 = p[3], P11 = p[4], P12 = p[5];
    const float P20 = p[6], P21 = p[7], P22 = p[8];
    const float c0 = p[9], c1 = p[10], c2 = p[11];

    const float fx = (float)(n & (Wdim - 1));
    const float fy = (float)(n >> 9);

    const float d = dsrc[b * NPIX + n];
    const float r0 = fx * P00 + fy * P10 + P20;
    const float r1 = fx * P01 + fy * P11 + P21;
    const float r2 = fx * P02 + fy * P12 + P22;
    const float uw = d * r0 + c0;
    const float vw = d * r1 + c1;
    const float w  = d * r2 + c2;

    const float denom = fmaxf(w, 0.0f) + 1e-12f;
    const float u = uw / denom;
    const float v = vw / denom;

    // Follow reference normalization literally.
    const float gx = u * (2.0f / (float)(Wdim - 1)) - 1.0f;
    const float gy = v * (2.0f / (float)(Hdim - 1)) - 1.0f;
    float xs = ((gx + 1.0f) * (float)Wdim - 1.0f) * 0.5f;
    float ys = ((gy + 1.0f) * (float)Hdim - 1.0f) * 0.5f;
    xs = fminf(fmaxf(xs, 0.0f), (float)(Wdim - 1));
    ys = fminf(fmaxf(ys, 0.0f), (float)(Hdim - 1));

    const float x0f = floorf(xs), y0f = floorf(ys);
    const float wx = xs - x0f, wy = ys - y0f;
    const int x0 = (int)x0f, y0 = (int)y0f;
    const int x1 = min(x0 + 1, Wdim - 1);
    const int y1 = min(y0 + 1, Hdim - 1);
    const float w00 = (1.0f - wx) * (1.0f - wy);
    const float w01 = wx * (1.0f - wy);
    const float w10 = (1.0f - wx) * wy;
    const float w11 = wx * wy;

    const int i00 = y0 * Wdim + x0, i01 = y0 * Wdim + x1;
    const int i10 = y1 * Wdim + x0, i11 = y1 * Wdim + x1;

    const float* db = ddst + b * NPIX;
    const float dsamp = db[i00] * w00 + db[i01] * w01 + db[i10] * w10 + db[i11] * w11;

    const float m = msrc[b * NPIX + n] ? 1.0f : 0.0f;
    float ld = fabsf(w - dsamp) * m;
    const float cm = (ld < CLAMP_T) ? 1.0f : 0.0f;
    ld *= cm;

    float lrgb = 0.0f;
    const float* cb  = cdst + b * 3 * NPIX;
    const float* c0p = csrc + b * 3 * NPIX;
    #pragma unroll
    for (int ch = 0; ch < 3; ch++) {
        const float* im = cb + ch * NPIX;
        const float s = im[i00] * w00 + im[i01] * w01 + im[i10] * w10 + im[i11] * w11;
        lrgb += fabsf(c0p[ch * NPIX + n] - s);
    }
    lrgb = lrgb * (1.0f / 3.0f) * m * cm;

    // wave32 reduce, then cross-wave via LDS (deterministic, atomic-free)
    #pragma unroll
    for (int off = 16; off > 0; off >>= 1) {
        ld   += __shfl_down(ld, off, 32);
        lrgb += __shfl_down(lrgb, off, 32);
    }
    __shared__ float sd[BLOCK / 32];
    __shared__ float sr[BLOCK / 32];
    const int lane = threadIdx.x & 31;
    const int wave = threadIdx.x >> 5;
    if (lane == 0) { sd[wave] = ld; sr[wave] = lrgb; }
    __syncthreads();
    if (threadIdx.x == 0) {
        float a = 0.f, r = 0.f;
        #pragma unroll
        for (int i = 0; i < BLOCK / 32; i++) { a += sd[i]; r += sr[i]; }
        const int chunk = b * CHUNKS_PER_BATCH + dir * PIXBLOCKS + blockIdx.x;
        pd[chunk] = a;
        prgb[chunk] = r;
    }
}

// ---------------------------------------------------------------------------
// Kernel 3: WMMA reduction.  D = A(16x4) * ones(4x16) + C  -> every D row j
// holds the running per-batch sum.  One wave per loss array (EXEC all-ones
// through the WMMA loop, divergence only after the last WMMA).
// ---------------------------------------------------------------------------
__global__ void rgbd_wmma_reduce(const float* __restrict__ pd,
                                 const float* __restrict__ prgb,
                                 float* __restrict__ out) {
    const float* arr = (blockIdx.x == 0) ? pd : prgb;
    float* o = out + blockIdx.x * BS;

    const int lane = threadIdx.x;           // 0..31
    const int row  = lane & 15;             // batch row (A matrix M)
    const int koff = (lane >> 4) * 2;       // lanes 0-15 -> K=0,1 ; lanes 16-31 -> K=2,3

    v8f acc = {};
    v2f bones;
    bones.x = 1.0f; bones.y = 1.0f;

    for (int k = 0; k < CHUNKS_PER_BATCH; k += 4) {
        const float* src = arr + row * CHUNKS_PER_BATCH + k + koff;
        v2f a;
        a.x = src[0];
        a.y = src[1];
        // D = A x ones + C : accumulates per-batch row sums in f32.
        acc = __builtin_amdgcn_wmma_f32_16x16x4_f32(
            /*neg_a=*/false, a, /*neg_b=*/false, bones,
            /*c_mod=*/(short)0, acc, /*reuse_a=*/false, /*reuse_b=*/false);
    }

    const float scale = 1.0f / (float)NPIX;  // mean over pixels; dirs already summed
    if (lane == 0) {
        #pragma unroll
        for (int i = 0; i < 8; i++) o[i] = acc[i] * scale;       // D[0..7][0]
    } else if (lane == 16) {
        #pragma unroll
        for (int i = 0; i < 8; i++) o[8 + i] = acc[i] * scale;   // D[8..15][0]
    }
}

// ---------------------------------------------------------------------------
extern "C" void kernel_launch(void* const* d_in, const int* in_sizes, int n_in,
                              void* d_out, int out_size, void* d_ws, size_t ws_size,
                              hipStream_t stream) {
    const float* depth0 = (const float*)d_in[0];
    const float* depth1 = (const float*)d_in[1];
    const float* R0     = (const float*)d_in[2];
    const float* t0     = (const float*)d_in[3];
    const float* R1     = (const float*)d_in[4];
    const float* t1     = (const float*)d_in[5];
    const float* rgb0   = (const float*)d_in[6];
    const float* rgb1   = (const float*)d_in[7];
    const unsigned char* mask0 = (const unsigned char*)d_in[8];
    const unsigned char* mask1 = (const unsigned char*)d_in[9];
    const float* K      = (const float*)d_in[10];

    float* ws     = (float*)d_ws;
    float* params = ws;                          // 2*16*12 = 384 floats
    float* pd     = ws + 512;                    // 16*2048 floats
    float* prgb   = pd + BS * CHUNKS_PER_BATCH;  // 16*2048 floats
    (void)in_sizes; (void)n_in; (void)out_size; (void)ws_size;

    rgbd_precompute<<<1, 32, 0, stream>>>(R0, t0, R1, t1, K, params);

    dim3 grid(PIXBLOCKS, BS, 2);
    rgbd_reproject_loss<<<grid, BLOCK, 0, stream>>>(depth0, depth1, rgb0, rgb1,
                                                    mask0, mask1, params, pd, prgb);

    rgbd_wmma_reduce<<<2, 32, 0, stream>>>(pd, prgb, (float*)d_out);
}